// ACTHalting_61194694033990
// MI455X (gfx1250) — compile-verified
//
#include <hip/hip_runtime.h>
#include <hip/hip_bf16.h>

typedef float v2f __attribute__((ext_vector_type(2)));
typedef float v8f __attribute__((ext_vector_type(8)));

#define ACT_B 256
#define ACT_T 64
#define ACT_D 2048
#define ACT_THRESHOLD 0.99f
#define ACT_EPSILON 0.01f

// ---------------------------------------------------------------------------
// Kernel 1: per-batch halting scan -> normalized weights + ponder cost.
// Tiny working set (256*64 floats); one thread per batch row is plenty.
// ---------------------------------------------------------------------------
__global__ void act_weights_kernel(const float* __restrict__ halt_probs, // [B,T] (trailing 1 dim flattened)
                                   const float* __restrict__ step_w,     // [B,T]
                                   float* __restrict__ weights_out,      // [B,T]
                                   float* __restrict__ ponder_out)       // [B]
{
    int b = blockIdx.x * blockDim.x + threadIdx.x;
    if (b >= ACT_B) return;

    const float* hp = halt_probs + (size_t)b * ACT_T;
    const float* sw = step_w     + (size_t)b * ACT_T;

    // Pass A: inclusive cumsum, find first index where cum >= threshold.
    float cum = 0.0f;
    int   hs = ACT_T - 1;
    bool  found = false;
    float cum_at = 0.0f, p_at = 0.0f;
    for (int t = 0; t < ACT_T; ++t) {
        float p = hp[t];
        cum += p;
        if (!found && cum >= ACT_THRESHOLD) {
            found = true; hs = t; cum_at = cum; p_at = p;
        }
    }
    if (!found) {                 // never exceeded: halt at T-1
        hs = ACT_T - 1; cum_at = cum; p_at = hp[ACT_T - 1];
    }
    float remaining = 1.0f - cum_at + p_at;

    // Pass B: step-weighted sum for normalization.
    float wsum = 0.0f;
    for (int t = 0; t < ACT_T; ++t) {
        float wt = (t < hs) ? hp[t] : ((t == hs) ? remaining : 0.0f);
        wsum += wt * sw[t];
    }
    float denom = fmaxf(wsum, ACT_EPSILON);   // clip(sum, EPS, None)

    // Pass C: write normalized weights, accumulate ponder cost.
    float ponder = 0.0f;
    for (int t = 0; t < ACT_T; ++t) {
        float wt = (t < hs) ? hp[t] : ((t == hs) ? remaining : 0.0f);
        wt = (wt * sw[t]) / denom;
        weights_out[(size_t)b * ACT_T + t] = wt;
        ponder += wt * (float)(t + 1);
    }
    ponder_out[b] = ponder;
}

// ---------------------------------------------------------------------------
// Kernel 2: final_output[b,d] = sum_t w[b,t] * O[b,t,d] via V_WMMA_F32_16X16X4_F32.
// One wave32 per (batch, 16-column d-tile). A = weights broadcast down rows,
// B = O tile (K=4 x N=16). All D rows equal; row 0 extracted from C VGPR0.
//
// A 16x4 f32 layout:  lanes 0-15 -> {K0,K1}, lanes 16-31 -> {K2,K3}
// B 4x16 f32 layout:  lane = column N (mod 16); lanes 0-15 -> {K0,K1},
//                     lanes 16-31 -> {K2,K3}
// C/D 16x16 f32:      VGPR0 lanes 0-15 = (M=0, N=lane)
// ---------------------------------------------------------------------------
__global__ void act_einsum_wmma(const float* __restrict__ outputs,  // [B,T,D]
                                const float* __restrict__ weights,  // [B,T] normalized
                                float* __restrict__ final_out)      // [B,D]
{
    const int TILES = ACT_D / 16;                 // 128 d-tiles per batch
    const int lane  = threadIdx.x & 31;
    const int wv    = threadIdx.x >> 5;
    const int task  = blockIdx.x * 8 + wv;        // 8 waves per block
    const int b     = task / TILES;
    const int tile  = task - b * TILES;
    const int d0    = tile << 4;

    const int col   = lane & 15;                  // N within tile
    const int koff  = (lane >> 4) << 1;           // 0 for lanes 0-15, 2 for 16-31

    const float* wp = weights + (size_t)b * ACT_T;
    const float* op = outputs + (size_t)b * ACT_T * ACT_D + (size_t)d0 + col;

    v8f c = {0.f, 0.f, 0.f, 0.f, 0.f, 0.f, 0.f, 0.f};

#pragma unroll
    for (int t0 = 0; t0 < ACT_T; t0 += 4) {
        v2f a, bb;
        // A[m,k] = w[t0+k] for every row m (broadcast).
        a.x = wp[t0 + koff];
        a.y = wp[t0 + koff + 1];
        // B[k,n] = O[t0+k, d0+n].
        const float* o = op + (size_t)(t0 + koff) * ACT_D;
        bb.x = o[0];
        bb.y = o[ACT_D];
        c = __builtin_amdgcn_wmma_f32_16x16x4_f32(
                /*neg_a=*/false, a, /*neg_b=*/false, bb,
                /*c_mod=*/(short)0, c, /*reuse_a=*/false, /*reuse_b=*/false);
    }

    // Row M=0 of D lives in c[0], lanes 0-15 (N = lane): contiguous 64B store.
    if (lane < 16) {
        final_out[(size_t)b * ACT_D + d0 + lane] = c[0];
    }
}

// ---------------------------------------------------------------------------
// Launch: d_out layout = final_output[256*2048] | ponder_cost[256] | weights[256*64]
// Kernel 2 consumes the weights section written by kernel 1 (same stream -> ordered).
// ---------------------------------------------------------------------------
extern "C" void kernel_launch(void* const* d_in, const int* in_sizes, int n_in,
                              void* d_out, int out_size, void* d_ws, size_t ws_size,
                              hipStream_t stream) {
    const float* halt_probs = (const float*)d_in[0];  // [B,T,1]
    const float* outputs    = (const float*)d_in[1];  // [B,T,D]
    const float* step_w     = (const float*)d_in[2];  // [B,T]

    float* out       = (float*)d_out;
    float* final_out = out;                                  // [B,D]
    float* ponder    = out + (size_t)ACT_B * ACT_D;          // [B]
    float* weights   = ponder + ACT_B;                       // [B,T]

    // 256 batch rows, 64 threads/block.
    act_weights_kernel<<<ACT_B / 64, 64, 0, stream>>>(halt_probs, step_w, weights, ponder);

    // 256 batches * 128 tiles = 32768 waves; 8 waves (256 threads) per block.
    act_einsum_wmma<<<(ACT_B * (ACT_D / 16)) / 8, 256, 0, stream>>>(outputs, weights, final_out);
}